// UpsampleRFFT_7636451852637
// MI455X (gfx1250) — compile-verified
//
#include <hip/hip_runtime.h>
#include <math.h>

// ---------------------------------------------------------------------------
// 2x band-limited upsample (zero-insert + ideal low-pass via rfft2) rewritten
// as three circulant 128x128x128 f32 matmuls per channel on gfx1250 wave32:
//
//   out[2i  ,2j  ] = X[i,j]
//   out[2i+1,2j  ] = Yh  = (C @ X      )[i,j]    C[p][q]  = w[(p-q) mod 128]
//   out[2i  ,2j+1] = Yw  = (X @ Ct     )[i,j]    Ct[p][q] = w[(q-p) mod 128]
//   out[2i+1,2j+1] = Yhw = (C @ X @ Ct )[i,j]
//   w[m] = 2*K[2m+1],  K[n] = sin(127*pi*n/256) / (256*sin(pi*n/256)), n odd
//
// Matrices used as WMMA B operands are stored K-pair interleaved:
//   Bi[(k>>1)*256 + n*2 + (k&1)] = B[k][n]
// so every A and B fragment is a single aligned b64 load into an even VGPR
// pair (no register re-pairing movs). Final writeback zips even/odd columns
// in registers -> fully coalesced b128 global stores.
// ---------------------------------------------------------------------------

typedef __attribute__((ext_vector_type(2))) float v2f;
typedef __attribute__((ext_vector_type(8))) float v8f;

#define N128 128
#define PI_D 3.14159265358979323846

__global__ __launch_bounds__(256) void build_filters_kernel(float* __restrict__ Wc,
                                                            float* __restrict__ WcTi) {
    __shared__ float w[N128];
    const int tid = threadIdx.x;
    if (tid < N128) {
        const int n = 2 * tid + 1;  // odd frequency index
        const double k = sin(127.0 * PI_D * n / 256.0) /
                         (256.0 * sin(PI_D * n / 256.0));
        w[tid] = (float)(2.0 * k);
    }
    __syncthreads();
    for (int idx = tid; idx < N128 * N128; idx += 256) {
        const int p = idx >> 7, q = idx & 127;
        Wc[idx] = w[(p - q) & 127];                            // row-major (A operand)
        WcTi[(p >> 1) * 256 + q * 2 + (p & 1)] = w[(q - p) & 127];  // interleaved (B operand)
    }
}

// One wave: 16x128 strip of D = A(128x128, row-major) @ B(128x128, interleaved).
// 8 simultaneous 16x16 f32 accumulators, 32 K-steps of wmma f32 16x16x4.
__device__ __forceinline__ void gemm128(const float* __restrict__ A,
                                        const float* __restrict__ Bi,
                                        float* __restrict__ DL,   // LDS sink, row-major
                                        int lane, int m0) {
    v8f acc[8];
#pragma unroll
    for (int t = 0; t < 8; ++t) acc[t] = (v8f)0.0f;

    const int lr = lane & 15;   // row of A tile / col of B tile (ISA 7.12.2)
    const int lh = lane >> 4;   // half-wave selects K pair

    const float* __restrict__ arow  = A + (m0 + lr) * N128 + 2 * lh;
    const float* __restrict__ bbase = Bi + lh * 256 + lr * 2;

    for (int k0 = 0; k0 < N128; k0 += 4) {
        const v2f a = *(const v2f*)(arow + k0);                 // A[m][k..k+1]
        const float* __restrict__ bp = bbase + (k0 >> 1) * 256;
#pragma unroll
        for (int t = 0; t < 8; ++t) {
            const v2f b = *(const v2f*)(bp + t * 32);           // B[k..k+1][n]
            acc[t] = __builtin_amdgcn_wmma_f32_16x16x4_f32(
                false, a, false, b, (short)0, acc[t], false, false);
        }
    }

    const int row0 = m0 + lh * 8;   // C/D layout: VGPR r -> row m0 + r + 8*lh
#pragma unroll
    for (int t = 0; t < 8; ++t) {
        const int col = t * 16 + lr;
#pragma unroll
        for (int r = 0; r < 8; ++r)
            DL[(row0 + r) * N128 + col] = acc[t][r];
    }
}

__global__ __launch_bounds__(256) void upsample2x_wmma_kernel(
    const float* __restrict__ x,      // [1024][128][128]
    float* __restrict__ out,          // [1024][256][256]
    const float* __restrict__ Wc,     // C, row-major
    const float* __restrict__ WcTi) { // C^T, K-pair interleaved
    extern __shared__ float smem[];
    float* Xs  = smem;                 // X row-major        (A for GEMM2)
    float* Xi  = smem + 16384;         // X interleaved      (B for GEMM1) -> later Yhw
    float* Yhs = smem + 2 * 16384;     // Yh = C@X           (A for GEMM3, output)
    float* Yws = smem + 3 * 16384;     // Yw = X@Ct          (output)

    const int ch = blockIdx.x;
    const float* __restrict__ xc = x + (size_t)ch * 16384;
    float* __restrict__ outc = out + (size_t)ch * 65536;
    const int tid = threadIdx.x;

    // Stage X: coalesced b128 loads -> row-major copy + interleaved copy.
#pragma unroll
    for (int v = 0; v < 16; ++v) {
        const int idx4 = v * 256 + tid;
        const float4 d = ((const float4*)xc)[idx4];
        ((float4*)Xs)[idx4] = d;
        const int base = idx4 * 4;
        const int i = base >> 7, j = base & 127;
        float* xi = Xi + (i >> 1) * 256 + (i & 1) + j * 2;
        xi[0] = d.x; xi[2] = d.y; xi[4] = d.z; xi[6] = d.w;
    }
    __syncthreads();

    const int lane = tid & 31;
    const int m0   = (tid >> 5) * 16;   // 8 waves -> 8 row strips

    gemm128(Wc,  Xi,   Yhs, lane, m0);  // Yh  = C  @ X
    __syncthreads();                    // Yhs ready; Xi free
    gemm128(Xs,  WcTi, Yws, lane, m0);  // Yw  = X  @ Ct
    gemm128(Yhs, WcTi, Xi,  lane, m0);  // Yhw = Yh @ Ct   (into Xi buffer)
    __syncthreads();

    // Writeback: zip even/odd columns in registers, b128 coalesced stores.
#pragma unroll 4
    for (int v = 0; v < 64; ++v) {
        const int fidx = v * 256 + tid;     // float4 index within channel
        const int u = fidx >> 6;            // output row 0..255
        const int q = fidx & 63;            // float4 within row
        const int i = u >> 1;
        const float* __restrict__ E = (u & 1) ? Yhs : Xs;   // even columns
        const float* __restrict__ O = (u & 1) ? Xi  : Yws;  // odd columns
        const v2f e = *(const v2f*)(E + i * N128 + q * 2);
        const v2f o = *(const v2f*)(O + i * N128 + q * 2);
        float4 r; r.x = e.x; r.y = o.x; r.z = e.y; r.w = o.y;
        ((float4*)outc)[u * 64 + q] = r;
    }
}

extern "C" void kernel_launch(void* const* d_in, const int* in_sizes, int n_in,
                              void* d_out, int out_size, void* d_ws, size_t ws_size,
                              hipStream_t stream) {
    const float* x = (const float*)d_in[0];   // 16*64*128*128 f32
    float* out  = (float*)d_out;              // 16*64*256*256 f32
    float* Wc   = (float*)d_ws;               // 128*128 f32
    float* WcTi = Wc + N128 * N128;           // 128*128 f32, interleaved

    build_filters_kernel<<<1, 256, 0, stream>>>(Wc, WcTi);

    const int channels = 16 * 64;
    const size_t shmem = 4u * N128 * N128 * sizeof(float);  // 256 KB of 320 KB WGP LDS
    upsample2x_wmma_kernel<<<channels, 256, shmem, stream>>>(x, out, Wc, WcTi);
}